// Res5ROIHeads_27462020891254
// MI455X (gfx1250) — compile-verified
//
#include <hip/hip_runtime.h>
#include <hip/hip_bf16.h>

// Problem sizes (fixed by the reference)
#define NB    16       // batch
#define MG    128      // gt boxes per image
#define NP    20000    // proposals per image
#define NUMC  80       // background class id
#define SAMP  512      // samples per image
#define FGMAX 128      // fg quota
#define T1    256      // match kernel block
#define T2    1024     // select kernel block (32 wave32 waves)
#define CH    ((NP + T2 - 1) / T2)   // 20 elements owned per thread
#define SENT  0xFFFFFFFFFFFFFFFFull

// ---------------------------------------------------------------------------
// CDNA5 async global->LDS staging (ASYNCcnt path). Builtin signatures (from
// clang diagnostics): b128 takes (AS1 v4i* src, AS3 v4i* dst, imm, imm),
// b32 takes (AS1 int* src, AS3 int* dst, imm, imm).
// ---------------------------------------------------------------------------
#if defined(__HIP_DEVICE_COMPILE__) && defined(__gfx1250__) && \
    __has_builtin(__builtin_amdgcn_global_load_async_to_lds_b128) && \
    __has_builtin(__builtin_amdgcn_global_load_async_to_lds_b32)
#define HAVE_ASYNC 1
typedef int v4i_async __attribute__((vector_size(4 * sizeof(int))));
#else
#define HAVE_ASYNC 0
#endif

__device__ __forceinline__ void async_b128(const void* g, void* l) {
#if HAVE_ASYNC
  __builtin_amdgcn_global_load_async_to_lds_b128(
      (__attribute__((address_space(1))) v4i_async*)g,
      (__attribute__((address_space(3))) v4i_async*)l, 0, 0);
#else
  const float* gf = (const float*)g;
  float* lf = (float*)l;
  lf[0] = gf[0]; lf[1] = gf[1]; lf[2] = gf[2]; lf[3] = gf[3];
#endif
}

__device__ __forceinline__ void async_b32(const void* g, void* l) {
#if HAVE_ASYNC
  __builtin_amdgcn_global_load_async_to_lds_b32(
      (__attribute__((address_space(1))) int*)g,
      (__attribute__((address_space(3))) int*)l, 0, 0);
#else
  *(int*)l = *(const int*)g;
#endif
}

__device__ __forceinline__ void wait_async0() {
#if HAVE_ASYNC
#if __has_builtin(__builtin_amdgcn_s_wait_asynccnt)
  __builtin_amdgcn_s_wait_asynccnt(0);
#else
  asm volatile("s_wait_asynccnt 0" ::: "memory");
#endif
#endif
}

// ---------------------------------------------------------------------------
// IoU match of one proposal against all 128 gt boxes in LDS.
// Strict '>' starting at m=0 reproduces jnp.argmax first-max semantics
// (all-zero IoU rows -> argmax 0, matched_val 0).
// ---------------------------------------------------------------------------
__device__ __forceinline__ void match_one(const float* s_gt, const int* s_gtc,
                                          float px1, float py1, float px2, float py2,
                                          float& mval, int& midx, int& mcls) {
  const float ap = (px2 - px1) * (py2 - py1);
  float best = -1.0f;
  int bi = 0;
#pragma unroll 4
  for (int m = 0; m < MG; ++m) {
    const float gx1 = s_gt[m * 4 + 0], gy1 = s_gt[m * 4 + 1];
    const float gx2 = s_gt[m * 4 + 2], gy2 = s_gt[m * 4 + 3];
    const float ag = (gx2 - gx1) * (gy2 - gy1);
    float iw = fminf(gx2, px2) - fmaxf(gx1, px1);
    float ih = fminf(gy2, py2) - fmaxf(gy1, py1);
    iw = fmaxf(iw, 0.0f);
    ih = fmaxf(ih, 0.0f);
    const float inter = iw * ih;
    const float iou = inter / (ag + ap - inter);
    if (iou > best) { best = iou; bi = m; }
  }
  mval = best;
  midx = bi;
  mcls = (best >= 0.5f) ? s_gtc[bi] : NUMC;
}

// ---------------------------------------------------------------------------
// Kernel 1: per-proposal match (wide parallelism). Stages gt table and a
// 256-proposal tile into LDS through the async engine.
// ---------------------------------------------------------------------------
__global__ __launch_bounds__(T1) void match_kernel(
    const float* __restrict__ gt_boxes, const int* __restrict__ gt_classes,
    const float* __restrict__ pr_boxes,
    float* __restrict__ ws_val, unsigned int* __restrict__ ws_pack) {
  __shared__ float s_gt[MG * 4];
  __shared__ int   s_gtc[MG];
  __shared__ float s_pr[T1 * 4];

  const int b = blockIdx.y;
  const int tid = threadIdx.x;
  const int n = blockIdx.x * T1 + tid;

  const float* gtb = gt_boxes + (size_t)b * MG * 4;
  const int*   gtc = gt_classes + (size_t)b * MG;
  const float* prb = pr_boxes + (size_t)b * NP * 4;

  if (tid < MG) {
    async_b128(gtb + tid * 4, s_gt + tid * 4);
    async_b32(gtc + tid, s_gtc + tid);
  }
  if (n < NP) async_b128(prb + (size_t)n * 4, s_pr + tid * 4);
  wait_async0();
  __syncthreads();

  if (n < NP) {
    float mv; int mi, mc;
    match_one(s_gt, s_gtc,
              s_pr[tid * 4 + 0], s_pr[tid * 4 + 1],
              s_pr[tid * 4 + 2], s_pr[tid * 4 + 3], mv, mi, mc);
    ws_val[(size_t)b * NP + n] = mv;
    ws_pack[(size_t)b * NP + n] = ((unsigned)mi << 8) | (unsigned)mc;
  }
}

// ---------------------------------------------------------------------------
// wave32 helpers for the selection reduction
// ---------------------------------------------------------------------------
__device__ __forceinline__ unsigned long long shflx_u64(unsigned long long v, int m) {
  unsigned int lo = (unsigned int)v;
  unsigned int hi = (unsigned int)(v >> 32);
  lo = (unsigned int)__shfl_xor((int)lo, m, 32);
  hi = (unsigned int)__shfl_xor((int)hi, m, 32);
  return ((unsigned long long)hi << 32) | lo;
}

__device__ __forceinline__ unsigned long long block_min_u64(
    unsigned long long v, unsigned long long* s_warp, unsigned long long* s_bc, int tid) {
  const int lane = tid & 31, wid = tid >> 5;
  for (int off = 16; off > 0; off >>= 1) {
    const unsigned long long o = shflx_u64(v, off);
    v = (o < v) ? o : v;
  }
  if (lane == 0) s_warp[wid] = v;
  __syncthreads();
  if (wid == 0) {
    unsigned long long w = s_warp[lane];   // exactly 32 waves -> 32 entries
    for (int off = 16; off > 0; off >>= 1) {
      const unsigned long long o = shflx_u64(w, off);
      w = (o < w) ? o : w;
    }
    if (lane == 0) *s_bc = w;
  }
  __syncthreads();
  return *s_bc;
}

// ---------------------------------------------------------------------------
// Kernel 2: per-image subsample + gather. One block per image.
// Sort key: (float_bits(key) << 32) | (n+1)  -- monotone for keys in [0,1),
// index term reproduces JAX's stable tie-break. Extraction is strictly
// increasing, so no "taken" flags are needed, and only the winning thread
// rescans its 20 owned elements each round.
// ---------------------------------------------------------------------------
__global__ __launch_bounds__(T2) void select_kernel(
    const float* __restrict__ gt_boxes, const int* __restrict__ gt_classes,
    const float* __restrict__ pr_boxes, const float* __restrict__ keys,
    const float* __restrict__ ws_val, const unsigned int* __restrict__ ws_pack,
    int has_ws, float* __restrict__ out) {
  __shared__ float s_gt[MG * 4];
  __shared__ int   s_gtc[MG];
  __shared__ unsigned long long s_warp[32];
  __shared__ unsigned long long s_bc;
  __shared__ unsigned int s_sel[SAMP];

  const int b = blockIdx.x;
  const int tid = threadIdx.x;

  const float* gtb = gt_boxes + (size_t)b * MG * 4;
  const int*   gtc = gt_classes + (size_t)b * MG;
  const float* prb = pr_boxes + (size_t)b * NP * 4;
  const float* keyb = keys + (size_t)b * NP;
  const float* wval = ws_val + (size_t)b * NP;
  const unsigned int* wpk = ws_pack + (size_t)b * NP;

  if (tid < MG) {
    async_b128(gtb + tid * 4, s_gt + tid * 4);
    async_b32(gtc + tid, s_gtc + tid);
  }
  wait_async0();
  __syncthreads();

  // fg bitmask for the CH (=20) proposals this thread owns (strided by T2).
  unsigned int fgmask = 0;
  for (int i = 0; i < CH; ++i) {
    const int n = i * T2 + tid;
    if (n < NP) {
      int fg;
      if (has_ws) {
        fg = ((wpk[n] & 0xFFu) != (unsigned)NUMC);
      } else {
        float mv; int mi, mc;
        match_one(s_gt, s_gtc, prb[n * 4 + 0], prb[n * 4 + 1],
                  prb[n * 4 + 2], prb[n * 4 + 3], mv, mi, mc);
        fg = (mc != NUMC);
      }
      fgmask |= (unsigned)fg << i;
    }
  }

  auto scan_min = [&](int want_fg, unsigned long long last) -> unsigned long long {
    unsigned long long mn = SENT;
    for (int i = 0; i < CH; ++i) {
      const int n = i * T2 + tid;
      if (n < NP && (int)((fgmask >> i) & 1u) == want_fg) {
        const unsigned long long c =
            ((unsigned long long)__float_as_uint(keyb[n]) << 32) | (unsigned)(n + 1);
        if (c > last && c < mn) mn = c;
      }
    }
    return mn;
  };

  // fg phase: up to 128 extractions in key order
  unsigned long long cur = scan_min(1, 0ull);
  int fg_take = 0;
  for (int r = 0; r < FGMAX; ++r) {
    const unsigned long long m = block_min_u64(cur, s_warp, &s_bc, tid);
    if (m == SENT) break;                         // uniform: fg stream exhausted
    if (tid == 0) s_sel[r] = (unsigned int)(m & 0xFFFFFFFFull) - 1u;
    if (cur == m) cur = scan_min(1, m);           // only the owner rescans
    fg_take = r + 1;
  }

  // bg phase: fill remaining 512 - fg_take slots
  const int nbg = SAMP - fg_take;
  cur = scan_min(0, 0ull);
  for (int r = 0; r < nbg; ++r) {
    const unsigned long long m = block_min_u64(cur, s_warp, &s_bc, tid);
    unsigned int sidx;
    if (m == SENT) {
      sidx = (unsigned int)r;                     // degenerate (no bg) fallback
    } else {
      sidx = (unsigned int)(m & 0xFFFFFFFFull) - 1u;
      if (cur == m) cur = scan_min(0, m);
    }
    if (tid == 0) s_sel[fg_take + r] = sidx;
  }
  __syncthreads();

  // gather outputs: [B,512,5] float_out, then classes, then indices (as f32)
  if (tid < SAMP) {
    const unsigned int s = s_sel[tid];
    float mv; int mi, mc;
    if (has_ws) {
      mv = wval[s];
      const unsigned int p = wpk[s];
      mi = (int)((p >> 8) & 0xFFu);
      mc = (int)(p & 0xFFu);
    } else {
      match_one(s_gt, s_gtc, prb[(size_t)s * 4 + 0], prb[(size_t)s * 4 + 1],
                prb[(size_t)s * 4 + 2], prb[(size_t)s * 4 + 3], mv, mi, mc);
    }
    float* fo = out + ((size_t)b * SAMP + tid) * 5;
    fo[0] = mv;
    fo[1] = s_gt[mi * 4 + 0];
    fo[2] = s_gt[mi * 4 + 1];
    fo[3] = s_gt[mi * 4 + 2];
    fo[4] = s_gt[mi * 4 + 3];
    out[(size_t)NB * SAMP * 5 + (size_t)b * SAMP + tid] = (float)mc;
    out[(size_t)NB * SAMP * 5 + (size_t)NB * SAMP + (size_t)b * SAMP + tid] = (float)(int)s;
  }
}

// ---------------------------------------------------------------------------
// Host launcher
// ---------------------------------------------------------------------------
extern "C" void kernel_launch(void* const* d_in, const int* in_sizes, int n_in,
                              void* d_out, int out_size, void* d_ws, size_t ws_size,
                              hipStream_t stream) {
  const float* gt_boxes   = (const float*)d_in[0];   // [16,128,4]
  const int*   gt_classes = (const int*)d_in[1];     // [16,128]
  const float* pr_boxes   = (const float*)d_in[2];   // [16,20000,4]
  const float* keys       = (const float*)d_in[3];   // [16,20000]
  float* out = (float*)d_out;                        // 40960 + 8192 + 8192 floats

  const size_t valBytes = (size_t)NB * NP * sizeof(float);
  const size_t pakBytes = (size_t)NB * NP * sizeof(unsigned int);
  const int has_ws = (ws_size >= valBytes + pakBytes) ? 1 : 0;
  float* ws_val = (float*)d_ws;
  unsigned int* ws_pack = (unsigned int*)((char*)d_ws + valBytes);

  if (has_ws) {
    dim3 g((NP + T1 - 1) / T1, NB);
    match_kernel<<<g, T1, 0, stream>>>(gt_boxes, gt_classes, pr_boxes, ws_val, ws_pack);
  }
  select_kernel<<<dim3(NB), T2, 0, stream>>>(gt_boxes, gt_classes, pr_boxes, keys,
                                             ws_val, ws_pack, has_ws, out);
}